// StaticGraphGNN_84018150244772
// MI455X (gfx1250) — compile-verified
//
#include <hip/hip_runtime.h>
#include <math.h>

// ---------------------------------------------------------------------------
// Problem constants (match reference)
// ---------------------------------------------------------------------------
#define GNN_N     20000
#define GNN_E     320000
#define GNN_D     256      // feature width everywhere (D_IN = HEADS*HID = D_OUT)
#define GNN_H1    4
#define GNN_C1    64
#define GNN_EPS   1e-5f
#define GNN_SLOPE 0.2f

typedef __attribute__((ext_vector_type(2))) float v2f;
typedef __attribute__((ext_vector_type(8))) float v8f;

// ---------------------------------------------------------------------------
// helpers
// ---------------------------------------------------------------------------
__device__ __forceinline__ float lrelu(float x) {
    return x > 0.0f ? x : GNN_SLOPE * x;
}

// float atomic max via the standard sign-split int trick (monotone, race-safe)
__device__ __forceinline__ void atomicMaxFloat(float* addr, float val) {
    if (val >= 0.0f) {
        atomicMax((int*)addr, __float_as_int(val));
    } else {
        atomicMin((unsigned int*)addr, __float_as_uint(val));
    }
}

// edge e -> (src,dst); self-loops appended after the E real edges
__device__ __forceinline__ void edge_sd(const int* __restrict__ ei, int e,
                                        int& s, int& d) {
    if (e < GNN_E) { s = ei[e]; d = ei[GNN_E + e]; }
    else           { s = e - GNN_E; d = s; }
}

// ---------------------------------------------------------------------------
// fill
// ---------------------------------------------------------------------------
__global__ void fill_kernel(float* __restrict__ p, int n, float v) {
    int t = blockIdx.x * blockDim.x + threadIdx.x;
    if (t < n) p[t] = v;
}

// ---------------------------------------------------------------------------
// WMMA f32 GEMM:  C[M,256] = A[M,256] * B[256,256]    (K = N = 256 fixed)
//
// Block = 256 threads = 8 waves.  Each block owns a 128-row x 64-col output
// panel: wave w computes rows [tmBase+w]*16..+15, all 64 cols (4 v8f accs).
// The 64x64 B chunk for the current K-slice is staged in LDS (16 KB) with
// coalesced float4 loads, so WMMA B operands come from ds_load instead of
// per-step global loads (B is reused by all 1250 row tiles; LDS kills the
// load->wait->wmma serialization seen in the previous build).
//
// V_WMMA_F32_16X16X4_F32 layouts:
//   A 16x4:  lane(0-15)=row M, VGPR0/1 = K {0,1} (lanes 0-15) / {2,3} (16-31)
//   B 4x16:  VGPR0/1 hold K rows {0,1}(lanes 0-15) / {2,3}(lanes 16-31), col=lane%16
//   C 16x16: VGPR v -> M = v + 8*(lane/16), N = lane%16
// ---------------------------------------------------------------------------
#define GEMM_KC 64

__global__ void gemm_wmma_f32(const float* __restrict__ A,
                              const float* __restrict__ B,
                              float* __restrict__ C, int M) {
    __shared__ float Bs[GEMM_KC * 64];          // 16 KB

    const int tilesN = 4;                        // 256 / 64
    const int tn     = blockIdx.x % tilesN;
    const int tmBase = (blockIdx.x / tilesN) * 8;
    const int wave   = threadIdx.x >> 5;
    const int tm     = tmBase + wave;
    const int lane   = threadIdx.x & 31;
    const int half   = lane >> 4;                // 0 or 1
    const int l16    = lane & 15;
    const bool active = (tm * 16 < M);           // wave-uniform

    const float* Arow = A + (size_t)(tm * 16 + l16) * 256;

    v8f acc0 = {}, acc1 = {}, acc2 = {}, acc3 = {};

    for (int kc = 0; kc < 256; kc += GEMM_KC) {
        __syncthreads();
        // cooperative stage of B[kc..kc+63, tn*64..+63] into LDS
#pragma unroll
        for (int j = 0; j < 4; j++) {
            int linear = j * 256 + threadIdx.x;  // 1024 float4 slots
            int row    = linear >> 4;
            int col4   = (linear & 15) << 2;
            *(float4*)(Bs + row * 64 + col4) =
                *(const float4*)(B + (size_t)(kc + row) * 256 + tn * 64 + col4);
        }
        __syncthreads();
        if (active) {
            if (kc + GEMM_KC < 256)
                __builtin_prefetch(Arow + kc + GEMM_KC, 0, 0);
#pragma unroll
            for (int k0 = 0; k0 < GEMM_KC; k0 += 4) {
                v2f a = *(const v2f*)(Arow + kc + k0 + 2 * half); // 8B-aligned
                const float* b0 = Bs + (k0 + 2 * half) * 64 + l16;
                const float* b1 = b0 + 64;
                v2f b;
                b.x = b0[0];  b.y = b1[0];
                acc0 = __builtin_amdgcn_wmma_f32_16x16x4_f32(
                    false, a, false, b, (short)0, acc0, false, false);
                b.x = b0[16]; b.y = b1[16];
                acc1 = __builtin_amdgcn_wmma_f32_16x16x4_f32(
                    false, a, false, b, (short)0, acc1, false, false);
                b.x = b0[32]; b.y = b1[32];
                acc2 = __builtin_amdgcn_wmma_f32_16x16x4_f32(
                    false, a, false, b, (short)0, acc2, false, false);
                b.x = b0[48]; b.y = b1[48];
                acc3 = __builtin_amdgcn_wmma_f32_16x16x4_f32(
                    false, a, false, b, (short)0, acc3, false, false);
            }
        }
    }

    if (!active) return;
    float* Cr = C + (size_t)(tm * 16 + 8 * half) * 256 + tn * 64 + l16;
#pragma unroll
    for (int v = 0; v < 8; v++) {
        Cr[(size_t)v * 256 + 0]  = acc0[v];
        Cr[(size_t)v * 256 + 16] = acc1[v];
        Cr[(size_t)v * 256 + 32] = acc2[v];
        Cr[(size_t)v * 256 + 48] = acc3[v];
    }
}

// ---------------------------------------------------------------------------
// per-(node,head) attention logits: es[i,h] = <h_row, a_src[h]>, ed likewise
// ---------------------------------------------------------------------------
__global__ void logits_kernel(const float* __restrict__ h,
                              const float* __restrict__ asrc,
                              const float* __restrict__ adst,
                              float* __restrict__ es, float* __restrict__ ed,
                              int n, int H, int C) {
    int t = blockIdx.x * blockDim.x + threadIdx.x;
    if (t >= n * H) return;
    int i  = t / H;
    int hd = t % H;
    const float* row = h + (size_t)i * H * C + hd * C;
    const float* as  = asrc + hd * C;
    const float* ad  = adst + hd * C;
    float sa = 0.0f, sb = 0.0f;
    for (int c = 0; c < C; c++) {
        float v = row[c];
        sa += v * as[c];
        sb += v * ad[c];
    }
    es[t] = sa;
    ed[t] = sb;
}

// ---------------------------------------------------------------------------
// segment-max of edge logits into m[dst,h]
// ---------------------------------------------------------------------------
__global__ void edge_max_kernel(const int* __restrict__ ei, int Etot,
                                const float* __restrict__ es,
                                const float* __restrict__ ed,
                                float* __restrict__ m, int H) {
    int t = blockIdx.x * blockDim.x + threadIdx.x;
    if (t >= Etot * H) return;
    int e = t / H, hd = t % H;
    int s, d;
    edge_sd(ei, e, s, d);
    float x = lrelu(es[s * H + hd] + ed[d * H + hd]);
    atomicMaxFloat(&m[d * H + hd], x);
}

// ---------------------------------------------------------------------------
// den[dst,h] += exp(e - m[dst,h])
// ---------------------------------------------------------------------------
__global__ void edge_den_kernel(const int* __restrict__ ei, int Etot,
                                const float* __restrict__ es,
                                const float* __restrict__ ed,
                                const float* __restrict__ m,
                                float* __restrict__ den, int H) {
    int t = blockIdx.x * blockDim.x + threadIdx.x;
    if (t >= Etot * H) return;
    int e = t / H, hd = t % H;
    int s, d;
    edge_sd(ei, e, s, d);
    float x = lrelu(es[s * H + hd] + ed[d * H + hd]);
    float w = __expf(x - m[d * H + hd]);
    atomicAdd(&den[d * H + hd], w);
}

// ---------------------------------------------------------------------------
// weighted scatter: agg[dst, :] += w(h) * h[src, :]
// 64 threads per edge, 4 channels each (float4 gather, 4 scalar atomics).
// ---------------------------------------------------------------------------
__global__ void edge_accum_kernel(const int* __restrict__ ei, int Etot,
                                  const float* __restrict__ es,
                                  const float* __restrict__ ed,
                                  const float* __restrict__ m,
                                  const float* __restrict__ hsrc,
                                  float* __restrict__ agg,
                                  int H, int C) {
    const int tpe = (H * C) >> 2;              // threads per edge (== 64)
    int t = blockIdx.x * blockDim.x + threadIdx.x;
    int e = t / tpe;
    if (e >= Etot) return;
    int q  = t % tpe;
    int c0 = q << 2;
    int hd = c0 / C;

    int s, d;
    edge_sd(ei, e, s, d);
    float x = lrelu(es[s * H + hd] + ed[d * H + hd]);
    float w = __expf(x - m[d * H + hd]);

    const float4 hv = *(const float4*)(hsrc + (size_t)s * H * C + c0);
    float* base = agg + (size_t)d * H * C + c0;
    atomicAdd(base + 0, w * hv.x);
    atomicAdd(base + 1, w * hv.y);
    atomicAdd(base + 2, w * hv.z);
    atomicAdd(base + 3, w * hv.w);
}

// ---------------------------------------------------------------------------
// layer-1 epilogue: v = agg/den + b1, LayerNorm(g,b), ReLU — in place.
// one 256-thread block per node, LDS tree reduction.
// ---------------------------------------------------------------------------
__global__ void finalize_ln_kernel(float* __restrict__ agg,
                                   const float* __restrict__ den,
                                   const float* __restrict__ bias,
                                   const float* __restrict__ g,
                                   const float* __restrict__ beta) {
    __shared__ float s1[GNN_D];
    __shared__ float s2[GNN_D];
    const int i = blockIdx.x;
    const int t = threadIdx.x;

    float v = agg[(size_t)i * GNN_D + t] / den[i * GNN_H1 + (t >> 6)] + bias[t];
    s1[t] = v;
    s2[t] = v * v;
    __syncthreads();
    for (int off = GNN_D / 2; off > 0; off >>= 1) {
        if (t < off) { s1[t] += s1[t + off]; s2[t] += s2[t + off]; }
        __syncthreads();
    }
    float mu  = s1[0] * (1.0f / GNN_D);
    float var = s2[0] * (1.0f / GNN_D) - mu * mu;
    float y = (v - mu) * rsqrtf(var + GNN_EPS) * g[t] + beta[t];
    agg[(size_t)i * GNN_D + t] = fmaxf(y, 0.0f);
}

// ---------------------------------------------------------------------------
// layer-2 epilogue: out = out/den + b2  (heads=1, mean over 1 head = identity)
// ---------------------------------------------------------------------------
__global__ void finalize2_kernel(float* __restrict__ out,
                                 const float* __restrict__ den,
                                 const float* __restrict__ bias) {
    int t = blockIdx.x * blockDim.x + threadIdx.x;
    if (t >= GNN_N * GNN_D) return;
    int i = t / GNN_D;
    int j = t % GNN_D;
    out[t] = out[t] / den[i] + bias[j];
}

// ---------------------------------------------------------------------------
// launcher
// ---------------------------------------------------------------------------
extern "C" void kernel_launch(void* const* d_in, const int* in_sizes, int n_in,
                              void* d_out, int out_size, void* d_ws, size_t ws_size,
                              hipStream_t stream) {
    (void)in_sizes; (void)n_in; (void)out_size; (void)ws_size;

    const float* x   = (const float*)d_in[0];
    const int*   ei  = (const int*)  d_in[1];   // [2,E] row-major (src row, dst row)
    const float* W1  = (const float*)d_in[2];
    const float* a1s = (const float*)d_in[3];
    const float* a1d = (const float*)d_in[4];
    const float* b1  = (const float*)d_in[5];
    const float* lng = (const float*)d_in[6];
    const float* lnb = (const float*)d_in[7];
    const float* W2  = (const float*)d_in[8];
    const float* a2s = (const float*)d_in[9];
    const float* a2d = (const float*)d_in[10];
    const float* b2  = (const float*)d_in[11];
    float*       out = (float*)d_out;

    const int N = GNN_N, D = GNN_D, H1 = GNN_H1, C1 = GNN_C1;
    const int Etot = GNN_E + GNN_N;

    // workspace carve-up (floats): total ~10.64M floats (~42.6 MB)
    float* ws   = (float*)d_ws;
    float* bufA = ws;                     // h1, later h2    [N*D]
    float* bufB = bufA + (size_t)N * D;   // agg1 -> hln     [N*D]
    float* es1  = bufB + (size_t)N * D;   // [N*H1]
    float* ed1  = es1 + N * H1;
    float* m1   = ed1 + N * H1;
    float* den1 = m1  + N * H1;
    float* es2  = den1 + N * H1;          // [N]
    float* ed2  = es2 + N;
    float* m2   = ed2 + N;
    float* den2 = m2  + N;

    const int TB = 256;
#define GRID(n) (((n) + TB - 1) / TB)
    const float NEG_INF = -__builtin_huge_valf();

    // GEMM grid: 4 col-panels x ceil(1250 row-tiles / 8 waves)
    const int gemmBlocks = 4 * ((N / 16 + 7) / 8);

    // ---- init accumulators -------------------------------------------------
    fill_kernel<<<GRID(N * H1), TB, 0, stream>>>(m1,   N * H1, NEG_INF);
    fill_kernel<<<GRID(N * H1), TB, 0, stream>>>(den1, N * H1, 0.0f);
    fill_kernel<<<GRID(N * D),  TB, 0, stream>>>(bufB, N * D,  0.0f);
    fill_kernel<<<GRID(N),      TB, 0, stream>>>(m2,   N,      NEG_INF);
    fill_kernel<<<GRID(N),      TB, 0, stream>>>(den2, N,      0.0f);
    fill_kernel<<<GRID(N * D),  TB, 0, stream>>>(out,  N * D,  0.0f);

    // ---- layer 1 -----------------------------------------------------------
    gemm_wmma_f32<<<gemmBlocks, TB, 0, stream>>>(x, W1, bufA, N);   // h1 = x@W1
    logits_kernel<<<GRID(N * H1), TB, 0, stream>>>(bufA, a1s, a1d, es1, ed1, N, H1, C1);
    edge_max_kernel<<<GRID(Etot * H1), TB, 0, stream>>>(ei, Etot, es1, ed1, m1, H1);
    edge_den_kernel<<<GRID(Etot * H1), TB, 0, stream>>>(ei, Etot, es1, ed1, m1, den1, H1);
    edge_accum_kernel<<<GRID(Etot * (H1 * C1 / 4)), TB, 0, stream>>>(
        ei, Etot, es1, ed1, m1, bufA, bufB, H1, C1);
    finalize_ln_kernel<<<N, GNN_D, 0, stream>>>(bufB, den1, b1, lng, lnb);

    // ---- layer 2 -----------------------------------------------------------
    gemm_wmma_f32<<<gemmBlocks, TB, 0, stream>>>(bufB, W2, bufA, N); // h2 = hln@W2
    logits_kernel<<<GRID(N), TB, 0, stream>>>(bufA, a2s, a2d, es2, ed2, N, 1, D);
    edge_max_kernel<<<GRID(Etot), TB, 0, stream>>>(ei, Etot, es2, ed2, m2, 1);
    edge_den_kernel<<<GRID(Etot), TB, 0, stream>>>(ei, Etot, es2, ed2, m2, den2, 1);
    edge_accum_kernel<<<GRID(Etot * (D / 4)), TB, 0, stream>>>(
        ei, Etot, es2, ed2, m2, bufA, out, 1, D);
    finalize2_kernel<<<GRID(N * D), TB, 0, stream>>>(out, den2, b2);
#undef GRID
}